// MOE_32693291057901
// MI455X (gfx1250) — compile-verified
//
#include <hip/hip_runtime.h>
#include <hip/hip_bf16.h>

#define INPUTS  2048
#define HIDDEN  2048
#define OUTPUTS 2048
#define EXPERTS 32

typedef float v2f __attribute__((ext_vector_type(2)));
typedef float v8f __attribute__((ext_vector_type(8)));

// ---------------------------------------------------------------------------
// Gate kernel: logits[e] = v . Gw[e,:] + Gb[e]; gate = softmax(logits).
// Also zeroes the accumulator buffer for the following stage (determinism
// under graph replay). One wave (32 lanes) per expert, 32 waves per block.
// ---------------------------------------------------------------------------
__global__ __launch_bounds__(1024) void moe_gate_kernel(
    const float* __restrict__ v,        // [K]
    const float* __restrict__ Gw,       // [EXPERTS, K]
    const float* __restrict__ Gb,       // [EXPERTS]
    float* __restrict__ gate_out,       // [EXPERTS]
    float* __restrict__ zero_buf,       // [zn] zeroed here
    int K, int zn)
{
    const int lane = threadIdx.x & 31;
    const int wave = threadIdx.x >> 5;   // expert id, 0..31

    // zero the next-stage accumulator
    for (int i = threadIdx.x; i < zn; i += blockDim.x) zero_buf[i] = 0.0f;

    // per-expert dot product
    const float* wrow = Gw + (size_t)wave * K;
    float s = 0.0f;
    for (int k = lane; k < K; k += 32) s += v[k] * wrow[k];
    #pragma unroll
    for (int off = 16; off > 0; off >>= 1) s += __shfl_xor(s, off, 32);

    __shared__ float logits[EXPERTS];
    if (lane == 0) logits[wave] = s + Gb[wave];
    __syncthreads();

    if (wave == 0) {
        float l = logits[lane];
        float m = l;
        #pragma unroll
        for (int off = 16; off > 0; off >>= 1) m = fmaxf(m, __shfl_xor(m, off, 32));
        float ex = __expf(l - m);
        float sum = ex;
        #pragma unroll
        for (int off = 16; off > 0; off >>= 1) sum += __shfl_xor(sum, off, 32);
        gate_out[lane] = ex / sum;
    }
}

// ---------------------------------------------------------------------------
// Stage kernel: out[n] += gate[e] * act( v . W[e,n,:] + b[e,n] )
// One wave handles a 16-neuron tile of one expert using V_WMMA_F32_16X16X4_F32:
//   A (16x4, src0) = broadcast of v[k..k+3] across all 16 M-rows (L0-resident)
//   B (4x16, src1) = W rows n0..n0+15, K-slice k..k+3 (HBM stream)
//   C row M=0, lanes 0..15 = y[n0..n0+15]
// A-layout (32-bit 16x4): lane L, vgpr p -> M=L%16, K=(L>>4)*2+p
// B-layout (32-bit 4x16): lane L, vgpr p -> N=L%16, K=(L>>4)*2+p
// Unroll 8 (32 K per outer iter) keeps ~16 loads in flight per wave so the
// 4096 streaming waves cover the ~16MB needed to saturate 23.3 TB/s HBM.
// ---------------------------------------------------------------------------
template <int RELU>
__global__ __launch_bounds__(256) void moe_stage_kernel(
    const float* __restrict__ v,        // [K] stage input
    const float* __restrict__ W,        // [EXPERTS, N, K]
    const float* __restrict__ b,        // [EXPERTS, N]
    const float* __restrict__ gate,     // [EXPERTS]
    float* __restrict__ out,            // [N] accumulator (pre-zeroed)
    int K, int N)
{
    const int lane = threadIdx.x & 31;
    const int wave = threadIdx.x >> 5;
    const int tile = blockIdx.x * (blockDim.x >> 5) + wave;
    const int tilesPerExpert = N >> 4;

    const int e  = tile / tilesPerExpert;
    const int n0 = (tile % tilesPerExpert) << 4;

    const int khalf = (lane >> 4) << 1;              // 0 for lanes 0-15, 2 for 16-31
    const float* wrow = W + ((size_t)e * N + n0 + (lane & 15)) * (size_t)K + khalf;
    const float* vptr = v + khalf;

    v8f c = {};
    for (int k = 0; k < K; k += 32) {
        // pull the weight stream ~4KB ahead (speculative; OOB is dropped)
        __builtin_prefetch(wrow + k + 1024, 0, 0);
        #pragma unroll
        for (int kk = 0; kk < 32; kk += 4) {
            v2f a  = *(const v2f*)(vptr + k + kk);   // broadcast of input vector
            v2f bm = *(const v2f*)(wrow + k + kk);   // streaming weights
            c = __builtin_amdgcn_wmma_f32_16x16x4_f32(
                    /*neg_a=*/false, a, /*neg_b=*/false, bm,
                    /*c_mod=*/(short)0, c, /*reuse_a=*/false, /*reuse_b=*/false);
        }
    }

    // C row M=0 lives in c[0] on lanes 0..15 (N = lane)
    if (lane < 16) {
        float y = c[0] + b[(size_t)e * N + n0 + lane];
        if (RELU) y = fmaxf(y, 0.0f);
        unsafeAtomicAdd(&out[n0 + lane], gate[e] * y);   // global_atomic_add_f32
    }
}

extern "C" void kernel_launch(void* const* d_in, const int* in_sizes, int n_in,
                              void* d_out, int out_size, void* d_ws, size_t ws_size,
                              hipStream_t stream) {
    const float* x   = (const float*)d_in[0];
    const float* G1w = (const float*)d_in[1];
    const float* G1b = (const float*)d_in[2];
    const float* G2w = (const float*)d_in[3];
    const float* G2b = (const float*)d_in[4];
    const float* G3w = (const float*)d_in[5];
    const float* G3b = (const float*)d_in[6];
    const float* W1  = (const float*)d_in[7];
    const float* b1  = (const float*)d_in[8];
    const float* W2  = (const float*)d_in[9];
    const float* b2  = (const float*)d_in[10];
    const float* W3  = (const float*)d_in[11];
    const float* b3  = (const float*)d_in[12];
    float* out = (float*)d_out;

    float* ws    = (float*)d_ws;
    float* gate1 = ws + 0;
    float* gate2 = ws + 32;
    float* gate3 = ws + 64;
    float* myh1  = ws + 96;
    float* myh2  = ws + 96 + HIDDEN;

    const int stageWaves  = EXPERTS * (HIDDEN / 16);   // 4096
    const int stageBlocks = stageWaves / 8;            // 512 blocks of 8 waves

    // Stage 1
    moe_gate_kernel<<<1, 1024, 0, stream>>>(x, G1w, G1b, gate1, myh1, INPUTS, HIDDEN);
    moe_stage_kernel<1><<<stageBlocks, 256, 0, stream>>>(x, W1, b1, gate1, myh1,
                                                         INPUTS, HIDDEN);
    // Stage 2
    moe_gate_kernel<<<1, 1024, 0, stream>>>(myh1, G2w, G2b, gate2, myh2, HIDDEN, HIDDEN);
    moe_stage_kernel<1><<<stageBlocks, 256, 0, stream>>>(myh1, W2, b2, gate2, myh2,
                                                         HIDDEN, HIDDEN);
    // Stage 3 (no relu)
    moe_gate_kernel<<<1, 1024, 0, stream>>>(myh2, G3w, G3b, gate3, out, HIDDEN, OUTPUTS);
    moe_stage_kernel<0><<<stageBlocks, 256, 0, stream>>>(myh2, W3, b3, gate3, out,
                                                         HIDDEN, OUTPUTS);
}